// GINModel_42528766165366
// MI455X (gfx1250) — compile-verified
//
#include <hip/hip_runtime.h>
#include <hip/hip_bf16.h>

#define N_NODES  50000
#define N_EDGES  600000
#define D_FEAT   128
#define H_MLP    256
#define N_LABELS 10

typedef __attribute__((ext_vector_type(2))) float v2f;
typedef __attribute__((ext_vector_type(8))) float v8f;

// ---------------------------------------------------------------------------
// bufA[i] = src[i]  (seed aggregation buffer with x, since eps == 0)
// ---------------------------------------------------------------------------
__global__ void copy_f4(const float4* __restrict__ src, float4* __restrict__ dst, int n4) {
    int i = blockIdx.x * blockDim.x + threadIdx.x;
    if (i < n4) dst[i] = src[i];
}

// ---------------------------------------------------------------------------
// Edge scatter: 32 lanes per edge, each lane moves 4 consecutive floats.
// agg[dst] += x[src] with native fp32 global atomics.
// ---------------------------------------------------------------------------
__global__ void scatter_add(const float* __restrict__ X,
                            const int* __restrict__ srcIdx,
                            const int* __restrict__ dstIdx,
                            float* __restrict__ agg) {
    int tid  = blockIdx.x * blockDim.x + threadIdx.x;
    int e    = tid >> 5;
    int lane = tid & 31;
    if (e >= N_EDGES) return;
    int s = srcIdx[e];
    int d = dstIdx[e];
    const float4 v = *(const float4*)(X + (size_t)s * D_FEAT + lane * 4);
    float* p = agg + (size_t)d * D_FEAT + lane * 4;
    unsafeAtomicAdd(p + 0, v.x);
    unsafeAtomicAdd(p + 1, v.y);
    unsafeAtomicAdd(p + 2, v.z);
    unsafeAtomicAdd(p + 3, v.w);
}

// ---------------------------------------------------------------------------
// out[N_NODES, NCOLS] = act(A[N_NODES, K] @ W[K, NCOLS] + bias)
// One wave per 16x16 output tile using V_WMMA_F32_16X16X4_F32.
//
// A-operand (16x4 f32, 2 VGPRs): lane L holds row M = L&15; half = L>>4
//   selects the K pair: {K = 2*half, 2*half+1}  -> one aligned b64 load.
// B-operand (4x16 f32, 2 VGPRs): lane L holds col N = L&15; half selects
//   rows {K = 2*half, 2*half+1}.
// C/D (16x16 f32, 8 VGPRs): VGPR j, lane L -> M = j + 8*(L>>4), N = L&15.
// 50000 % 16 == 0 so there is no divergence: EXEC is all-1s for WMMA.
// ---------------------------------------------------------------------------
template <int K, int NCOLS, bool RELU>
__global__ void gemm_wmma(const float* __restrict__ A,
                          const float* __restrict__ W,
                          const float* __restrict__ bias,
                          float* __restrict__ out) {
    const int wave = blockIdx.x * (blockDim.x >> 5) + (threadIdx.x >> 5);
    const int lane = threadIdx.x & 31;
    constexpr int NT = NCOLS / 16;
    const int mt   = wave / NT;
    const int nt   = wave % NT;
    const int m    = lane & 15;       // also the N index for B/C/D operands
    const int half = lane >> 4;

    const int row = mt * 16 + m;
    const int col = nt * 16 + m;

    const float* __restrict__ arow = A + (size_t)row * K;

    v8f acc = {};
#pragma unroll
    for (int kk = 0; kk < K; kk += 4) {
        const int ka = kk + half * 2;
        // A: two consecutive K values for this lane's row (8B aligned)
        v2f a = *(const v2f*)(arow + ka);
        // B: rows ka and ka+1 of W at this lane's column
        v2f b;
        b.x = W[(size_t)ka * NCOLS + col];
        b.y = W[(size_t)(ka + 1) * NCOLS + col];
        acc = __builtin_amdgcn_wmma_f32_16x16x4_f32(
            /*neg_a=*/false, a, /*neg_b=*/false, b,
            /*c_mod=*/(short)0, acc, /*reuse_a=*/false, /*reuse_b=*/false);
    }

    const float bv = bias[col];
#pragma unroll
    for (int j = 0; j < 8; ++j) {
        float v = acc[j] + bv;
        if (RELU) v = v > 0.0f ? v : 0.0f;
        out[(size_t)(mt * 16 + j + 8 * half) * NCOLS + col] = v;
    }
}

// ---------------------------------------------------------------------------
// Final tiny dense: out[node, l] = H[node,:] @ Wm2[:, l] + bm2[l]
// 16 threads per node (10 active labels), Wm2 stays L0/L2 resident.
// ---------------------------------------------------------------------------
__global__ void final_dense(const float* __restrict__ H,
                            const float* __restrict__ Wm2,
                            const float* __restrict__ bm2,
                            float* __restrict__ out) {
    int tid  = blockIdx.x * blockDim.x + threadIdx.x;
    int node = tid >> 4;
    int l    = tid & 15;
    if (node >= N_NODES || l >= N_LABELS) return;
    const float* __restrict__ h = H + (size_t)node * H_MLP;
    float acc = bm2[l];
#pragma unroll 8
    for (int k = 0; k < H_MLP; ++k)
        acc = fmaf(h[k], Wm2[k * N_LABELS + l], acc);
    out[(size_t)node * N_LABELS + l] = acc;
}

// ---------------------------------------------------------------------------
extern "C" void kernel_launch(void* const* d_in, const int* in_sizes, int n_in,
                              void* d_out, int out_size, void* d_ws, size_t ws_size,
                              hipStream_t stream) {
    const float* x    = (const float*)d_in[0];
    const int*   eidx = (const int*)d_in[1];     // [2, E]: row0 = src, row1 = dst
    const float* W0   = (const float*)d_in[2];
    const float* b0   = (const float*)d_in[3];
    const float* W1   = (const float*)d_in[4];
    const float* b1   = (const float*)d_in[5];
    const float* W2   = (const float*)d_in[6];
    const float* b2   = (const float*)d_in[7];
    const float* Wm1  = (const float*)d_in[8];
    const float* bm1  = (const float*)d_in[9];
    const float* Wm2  = (const float*)d_in[10];
    const float* bm2  = (const float*)d_in[11];
    float*       out  = (float*)d_out;

    const int* srcIdx = eidx;
    const int* dstIdx = eidx + N_EDGES;

    float* bufA = (float*)d_ws;                          // [N, 128] aggregation
    float* bufB = bufA + (size_t)N_NODES * D_FEAT;       // [N, 128] hidden
    float* bufC = bufB + (size_t)N_NODES * D_FEAT;       // [N, 256] mlp hidden

    const int n4        = N_NODES * D_FEAT / 4;          // 1.6M float4
    const int copyBlks  = (n4 + 255) / 256;
    const int scatBlks  = (N_EDGES * 32) / 256;          // 75000
    const int gemmBlks128 = (N_NODES / 16) * (D_FEAT / 16) / 8;   // 3125
    const int gemmBlks256 = (N_NODES / 16) * (H_MLP / 16) / 8;    // 6250
    const int finBlks   = (N_NODES * 16 + 255) / 256;

    // ---- Layer 0: agg = x + scatter(x); h0 = relu(agg @ W0 + b0) ----
    copy_f4<<<copyBlks, 256, 0, stream>>>((const float4*)x, (float4*)bufA, n4);
    scatter_add<<<scatBlks, 256, 0, stream>>>(x, srcIdx, dstIdx, bufA);
    gemm_wmma<D_FEAT, D_FEAT, true><<<gemmBlks128, 256, 0, stream>>>(bufA, W0, b0, bufB);

    // ---- Layer 1 ----
    copy_f4<<<copyBlks, 256, 0, stream>>>((const float4*)bufB, (float4*)bufA, n4);
    scatter_add<<<scatBlks, 256, 0, stream>>>(bufB, srcIdx, dstIdx, bufA);
    gemm_wmma<D_FEAT, D_FEAT, true><<<gemmBlks128, 256, 0, stream>>>(bufA, W1, b1, bufB);

    // ---- Layer 2 ----
    copy_f4<<<copyBlks, 256, 0, stream>>>((const float4*)bufB, (float4*)bufA, n4);
    scatter_add<<<scatBlks, 256, 0, stream>>>(bufB, srcIdx, dstIdx, bufA);
    gemm_wmma<D_FEAT, D_FEAT, true><<<gemmBlks128, 256, 0, stream>>>(bufA, W2, b2, bufB);

    // ---- Head: hid = relu(h2 @ Wm1 + bm1); out = hid @ Wm2 + bm2 ----
    gemm_wmma<D_FEAT, H_MLP, true><<<gemmBlks256, 256, 0, stream>>>(bufB, Wm1, bm1, bufC);
    final_dense<<<finBlks, 256, 0, stream>>>(bufC, Wm2, bm2, out);
}